// SpectralConv2d_memory_37898791420615
// MI455X (gfx1250) — compile-verified
//
#include <hip/hip_runtime.h>

typedef __attribute__((ext_vector_type(2))) float v2f;
typedef __attribute__((ext_vector_type(8))) float v8f;

#define TWO_PI 6.283185307179586f

__device__ __forceinline__ v8f wmma4(v2f a, v2f b, v8f c) {
  // V_WMMA_F32_16X16X4_F32: (neg_a, A, neg_b, B, c_mod, C, reuse_a, reuse_b)
  return __builtin_amdgcn_wmma_f32_16x16x4_f32(false, a, false, b, (short)0, c,
                                               false, false);
}

// ---------------------------------------------------------------------------
// K0: twiddle tables in workspace (recomputed every launch; deterministic)
//   TW1 [256w][48]  : fwd w-DFT  cols 2kx=cos, 2kx+1=-sin, 0 for n>=34
//   TW2 [33m][256h][2] : e^{-2pi i h (m-16)/256}  (cos,sin of -theta)
//   TW3 [256h][33m][2] : e^{+2pi i h (m-16)/256}
//   TW4 [48n][256w] : inv w-DFT  n=2kx: a*cos, n=2kx+1: -a*sin (a=1 if kx==0 else 2)
// ---------------------------------------------------------------------------
__global__ __launch_bounds__(256) void twiddle_init(float* __restrict__ ws) {
  int t = blockIdx.x * 256 + threadIdx.x;
  if (t < 12288) {
    int w = t / 48, n = t % 48;
    float v = 0.f;
    if (n < 34) {
      int kx = n >> 1;
      float ang = TWO_PI * (float)((w * kx) & 255) * (1.0f / 256.0f);
      v = (n & 1) ? -__sinf(ang) : __cosf(ang);
    }
    ws[t] = v;
  } else if (t < 29184) {
    int t2 = t - 12288;
    int m = t2 / 512, rem = t2 % 512;
    int h = rem >> 1, which = rem & 1;
    int p = ((h * (m - 16)) % 256 + 256) & 255;
    float ang = -TWO_PI * (float)p * (1.0f / 256.0f);
    ws[t] = which ? __sinf(ang) : __cosf(ang);
  } else if (t < 46080) {
    int t3 = t - 29184;
    int h = t3 / 66, rem = t3 % 66;
    int m = rem >> 1, which = rem & 1;
    int p = ((h * (m - 16)) % 256 + 256) & 255;
    float ang = TWO_PI * (float)p * (1.0f / 256.0f);
    ws[t] = which ? __sinf(ang) : __cosf(ang);
  } else if (t < 58368) {
    int t4 = t - 46080;
    int n = t4 / 256, w = t4 % 256;
    float v = 0.f;
    if (n < 34) {
      int kx = n >> 1;
      float a = (kx == 0) ? 1.f : 2.f;
      float ang = TWO_PI * (float)((w * kx) & 255) * (1.0f / 256.0f);
      v = (n & 1) ? -a * __sinf(ang) : a * __cosf(ang);
    }
    ws[t] = v;
  }
}

// ---------------------------------------------------------------------------
// K1: forward. One WG per (b,ci) image.
//   GEMM  T[256h][48] = X[256h][256w] @ TW1[256w][48]   (f32 WMMA, LDS out)
//   then 33-point ky-DFT:  G[m][kx] = (1/65536) * sum_h T[h][kx] * TW2[m][h]
// ---------------------------------------------------------------------------
__global__ __launch_bounds__(256) void fwd_dft(const float* __restrict__ x,
                                               const float* __restrict__ tw1,
                                               const float* __restrict__ tw2,
                                               float* __restrict__ G) {
  __shared__ float T2[256 * 48];
  const int tid = threadIdx.x;
  const int wv = tid >> 5;
  const int lane = tid & 31;
  const int half = lane >> 4;
  const int l16 = lane & 15;
  const float* xp = x + (size_t)blockIdx.x * 65536u;

  for (int job = wv; job < 48; job += 8) {
    const int mt = job / 3, nt = job % 3;
    const int row = mt * 16 + l16;
    const int col = nt * 16 + l16;
    v8f acc = {0.f, 0.f, 0.f, 0.f, 0.f, 0.f, 0.f, 0.f};
    for (int k0 = 0; k0 < 256; k0 += 4) {
      const int kk = k0 + 2 * half;
      v2f a = *(const v2f*)(xp + row * 256 + kk);
      v2f b;
      b.x = tw1[kk * 48 + col];
      b.y = tw1[(kk + 1) * 48 + col];
      acc = wmma4(a, b, acc);
    }
    const int rbase = mt * 16 + half * 8;
#pragma unroll
    for (int v = 0; v < 8; ++v) T2[(rbase + v) * 48 + col] = acc[v];
  }
  __syncthreads();

  for (int t = tid; t < 561; t += 256) {
    const int m = t / 17, kx = t % 17;
    const float* tw = tw2 + m * 512;
    float gre = 0.f, gim = 0.f;
    for (int h = 0; h < 256; ++h) {
      float tre = T2[h * 48 + 2 * kx];
      float tim = T2[h * 48 + 2 * kx + 1];
      float c = tw[2 * h], s = tw[2 * h + 1];
      gre += tre * c - tim * s;
      gim += tre * s + tim * c;
    }
    float* gp = G + ((size_t)t * 512 + blockIdx.x) * 2;
    gp[0] = gre * (1.0f / 65536.0f);
    gp[1] = gim * (1.0f / 65536.0f);
  }
}

// ---------------------------------------------------------------------------
// K2: channel mix per mode t=(m,kx):
//   R[b,co] = sum_ci G[b,ci] * conj(W[ci,co,m,16-kx])
// M=co (64), K=ci (64), N = 16 columns = [Gre(b=0..7) | Gim(b=0..7)].
// P1 = Wre^T@B, P2 = Wim^T@B; combine halves across lanes with shfl_xor(8):
//   Re = P1(lo) + P2(hi),  Im = P1(hi) - P2(lo)
// ---------------------------------------------------------------------------
__global__ __launch_bounds__(128) void chan_mix(const float* __restrict__ wt,
                                                const float* __restrict__ G,
                                                float* __restrict__ R) {
  __shared__ float Are[64 * 64];
  __shared__ float Aim[64 * 64];
  const int t = blockIdx.x;
  const int m = t / 17, kx = t % 17;
  const int tid = threadIdx.x;
  const int wv = tid >> 5;
  const int lane = tid & 31;
  const int half = lane >> 4;
  const int l16 = lane & 15;

  for (int i = tid; i < 4096; i += 128) {
    const int ci = i >> 6, co = i & 63;
    const size_t wofs = ((((size_t)ci * 64 + co) * 33 + m) * 17 + (16 - kx)) * 2;
    Are[co * 64 + ci] = wt[wofs];
    Aim[co * 64 + ci] = wt[wofs + 1];
  }
  __syncthreads();

  const float* Gp = G + (size_t)t * 1024;
  const int col = l16;
  const int bb = col & 7;     // batch index
  const int part = col >> 3;  // 0 = real half, 1 = imag half
  const int row = wv * 16 + l16;

  v8f p1 = {0.f, 0.f, 0.f, 0.f, 0.f, 0.f, 0.f, 0.f};
  v8f p2 = {0.f, 0.f, 0.f, 0.f, 0.f, 0.f, 0.f, 0.f};
  for (int k0 = 0; k0 < 64; k0 += 4) {
    const int kk = k0 + 2 * half;
    v2f ar, ai, b;
    ar.x = Are[row * 64 + kk];
    ar.y = Are[row * 64 + kk + 1];
    ai.x = Aim[row * 64 + kk];
    ai.y = Aim[row * 64 + kk + 1];
    b.x = Gp[(bb * 64 + kk) * 2 + part];
    b.y = Gp[(bb * 64 + kk + 1) * 2 + part];
    p1 = wmma4(ar, b, p1);
    p2 = wmma4(ai, b, p2);
  }
  const int cobase = wv * 16 + half * 8;
#pragma unroll
  for (int v = 0; v < 8; ++v) {
    float sw = __shfl_xor(p2[v], 8, 32);
    float res = (col < 8) ? (p1[v] + sw) : (p1[v] - sw);
    const int co = cobase + v;
    R[(((size_t)bb * 64 + co) * 561 + t) * 2 + part] = res;
  }
}

// ---------------------------------------------------------------------------
// K3: inverse. One WG per (b,co) image.
//   Z[h][kx] = sum_m R[m][kx] * TW3[h][m]   (33-point ky-iDFT, into LDS)
//   out[h][w] = sum_n Z2[h][n] * TW4[n][w] + bias   (f32 WMMA GEMM)
// TW4 row 1 is all-zero -> Im(Z[h,0]) dropped, matching irfft bin-0 behavior.
// ---------------------------------------------------------------------------
__global__ __launch_bounds__(256) void inv_dft(const float* __restrict__ R,
                                               const float* __restrict__ tw3,
                                               const float* __restrict__ tw4,
                                               const float* __restrict__ bias,
                                               float* __restrict__ out) {
  __shared__ float Z2[256 * 48];
  __shared__ float Rs[561 * 2];
  const int tid = threadIdx.x;
  const int wv = tid >> 5;
  const int lane = tid & 31;
  const int half = lane >> 4;
  const int l16 = lane & 15;
  const int img = blockIdx.x;

  for (int i = tid; i < 1122; i += 256) Rs[i] = R[(size_t)img * 1122 + i];
  __syncthreads();

  for (int idx = tid; idx < 256 * 17; idx += 256) {
    const int h = idx / 17, kx = idx % 17;
    float zre = 0.f, zim = 0.f;
    for (int mm = 0; mm < 33; ++mm) {
      const float rre = Rs[(mm * 17 + kx) * 2];
      const float rim = Rs[(mm * 17 + kx) * 2 + 1];
      const float c = tw3[(h * 33 + mm) * 2];
      const float s = tw3[(h * 33 + mm) * 2 + 1];
      zre += rre * c - rim * s;
      zim += rre * s + rim * c;
    }
    Z2[h * 48 + 2 * kx] = zre;
    Z2[h * 48 + 2 * kx + 1] = zim;
  }
  for (int idx = tid; idx < 256 * 14; idx += 256) {
    const int h = idx / 14, n = 34 + idx % 14;
    Z2[h * 48 + n] = 0.f;
  }
  __syncthreads();

  const float bv = bias[img & 63];
  float* op = out + (size_t)img * 65536u;
  for (int job = wv; job < 256; job += 8) {
    const int mt = job >> 4, nt = job & 15;
    const int row = mt * 16 + l16;
    const int col = nt * 16 + l16;
    v8f acc = {0.f, 0.f, 0.f, 0.f, 0.f, 0.f, 0.f, 0.f};
    for (int k0 = 0; k0 < 48; k0 += 4) {
      const int kk = k0 + 2 * half;
      v2f a, b;
      a.x = Z2[row * 48 + kk];
      a.y = Z2[row * 48 + kk + 1];
      b.x = tw4[kk * 256 + col];
      b.y = tw4[(kk + 1) * 256 + col];
      acc = wmma4(a, b, acc);
    }
    const int rbase = mt * 16 + half * 8;
#pragma unroll
    for (int v = 0; v < 8; ++v) op[(rbase + v) * 256 + col] = acc[v] + bv;
  }
}

extern "C" void kernel_launch(void* const* d_in, const int* in_sizes, int n_in,
                              void* d_out, int out_size, void* d_ws,
                              size_t ws_size, hipStream_t stream) {
  const float* x = (const float*)d_in[0];
  const float* wt = (const float*)d_in[1];    // complex64 interleaved
  const float* bias = (const float*)d_in[2];
  float* out = (float*)d_out;
  float* ws = (float*)d_ws;

  float* TW1 = ws;             // 12288 f32
  float* TW2 = ws + 12288;     // 16896 f32
  float* TW3 = ws + 29184;     // 16896 f32
  float* TW4 = ws + 46080;     // 12288 f32
  float* G = ws + 58368;       // 561*512*2 = 574464 f32
  float* R = ws + 632832;      // 512*561*2 = 574464 f32  (total ~4.6 MB)

  twiddle_init<<<228, 256, 0, stream>>>(ws);
  fwd_dft<<<512, 256, 0, stream>>>(x, TW1, TW2, G);
  chan_mix<<<561, 128, 0, stream>>>(wt, G, R);
  inv_dft<<<512, 256, 0, stream>>>(R, TW3, TW4, bias, out);
}